// NetConv_73839077753051
// MI455X (gfx1250) — compile-verified
//
#include <hip/hip_runtime.h>
#include <hip/hip_bf16.h>

typedef __attribute__((ext_vector_type(2))) float v2f;
typedef __attribute__((ext_vector_type(8))) float v8f;

#define IN_C   32
#define OUT_C  32
#define EDGE_F 16

#define WLDS_FLOATS (16 * 1024)   // W_edge re-swizzled for WMMA A operands (64 KB)
#define XS_STRIDE   36            // padded row stride for x-tile (bank-conflict free)
#define XS_FLOATS   (16 * XS_STRIDE)
#define WAVES_PER_BLOCK 8

__global__ void nnconv_zero_kernel(float* __restrict__ p, int n) {
    int i = blockIdx.x * blockDim.x + threadIdx.x;
    if (i < n) p[i] = 0.0f;
}

__device__ __forceinline__ float relu1(float v) {
    // single v_med3_f32: med3(x, 0, +inf) == max(x, 0) for non-NaN x
    return __builtin_amdgcn_fmed3f(v, 0.0f, __builtin_inff());
}

// One wave processes 16 edges per iteration.
// t[16 flat-cols x 16 edges] = W'_tile(16x16) x Attr'(16x16) via 4x wmma_f32_16x16x4_f32,
// C initialized with b_edge (bias folded in), then relu and FMA with x[src][i].
__global__ __launch_bounds__(256) void nnconv_edge_kernel(
    const float* __restrict__ x,
    const long long* __restrict__ edge_index,  // [2, E] int64: row0=src, row1=dst
    const float* __restrict__ edge_attr,       // [E, 16]
    const float* __restrict__ W_edge,          // [16, 1024]
    const float* __restrict__ b_edge,          // [1024]
    float* __restrict__ summed,                // [N, 32] (atomic accum)
    float* __restrict__ cnt,                   // [N]     (atomic accum)
    int E, int G)
{
    extern __shared__ float smem[];
    float* wlds = smem;
    const int tid  = threadIdx.x;
    const int lane = tid & 31;
    const int wave = tid >> 5;
    float* xs = smem + WLDS_FLOATS + wave * XS_FLOATS;

    // Stage W_edge into LDS pre-swizzled into WMMA A-operand order:
    // wlds[((tile*4+kks)*32 + lane)*2 + j] = W_edge[kks*4 + 2*(lane>>4) + j][tile*16 + (lane&15)]
    for (int idx = tid; idx < WLDS_FLOATS; idx += blockDim.x) {
        int j    = idx & 1;
        int ln   = (idx >> 1) & 31;
        int kks  = (idx >> 6) & 3;
        int tile = idx >> 8;
        int m    = ln & 15;
        int h2   = ln >> 4;
        int k    = kks * 4 + 2 * h2 + j;
        int col  = tile * 16 + m;
        wlds[idx] = W_edge[k * 1024 + col];
    }
    __syncthreads();

    const int e_lane = lane & 15;   // which edge of the 16-group this lane serves
    const int hi     = lane >> 4;   // lane half
    const int stride = gridDim.x * WAVES_PER_BLOCK;

    for (int g = blockIdx.x * WAVES_PER_BLOCK + wave; g < G; g += stride) {
        const int e0 = g * 16;
        int e = e0 + e_lane;
        const bool valid = (e < E);
        if (!valid) e = E - 1;                  // clamp; results discarded at write
        const int srcn = (int)edge_index[e];
        const int dstn = (int)edge_index[(size_t)E + e];

        // Stage x[src] rows (16 edges x 32 ch) into padded LDS tile.
        {
            const float4* xr = (const float4*)(x + (size_t)srcn * IN_C + 16 * hi);
            float* xd = xs + e_lane * XS_STRIDE + 16 * hi;
            float4 a0 = xr[0], a1 = xr[1], a2 = xr[2], a3 = xr[3];
            *(float4*)(xd + 0)  = a0;
            *(float4*)(xd + 4)  = a1;
            *(float4*)(xd + 8)  = a2;
            *(float4*)(xd + 12) = a3;
        }
        // Make cross-lane LDS writes visible within the wave (lockstep wave32).
        asm volatile("s_wait_dscnt 0" ::: "memory");

        // B operands: Attr'[K][edge]; per lane: edge_attr[e][kks*4 + 2*hi + {0,1}]
        v2f bop[4];
        {
            const float* ear = edge_attr + (size_t)e * EDGE_F + 2 * hi;
            #pragma unroll
            for (int kks = 0; kks < 4; ++kks)
                bop[kks] = *(const v2f*)(ear + 4 * kks);
        }

        v8f acc0 = {0.f,0.f,0.f,0.f,0.f,0.f,0.f,0.f};
        v8f acc1 = acc0;

        for (int i = 0; i < IN_C; ++i) {
            const float xv = xs[e_lane * XS_STRIDE + i];   // LDS broadcast per channel
            #pragma unroll
            for (int h = 0; h < 2; ++h) {
                const int tile = i * 2 + h;  // flat cols [tile*16, tile*16+16) = (i, o=16h..16h+15)
                // C init = b_edge slice (bias folded into accumulator before relu)
                const float4* bq = (const float4*)(b_edge + tile * 16 + 8 * hi);
                float4 bA = bq[0], bB = bq[1];
                v8f c = {bA.x, bA.y, bA.z, bA.w, bB.x, bB.y, bB.z, bB.w};
                const float* wl = wlds + tile * 256 + lane * 2;
                #pragma unroll
                for (int kks = 0; kks < 4; ++kks) {
                    v2f a = *(const v2f*)(wl + kks * 64);
                    c = __builtin_amdgcn_wmma_f32_16x16x4_f32(
                            /*neg_a=*/false, a, /*neg_b=*/false, bop[kks],
                            /*c_mod=*/(short)0, c, /*reuse_a=*/false, /*reuse_b=*/false);
                }
                #pragma unroll
                for (int r = 0; r < 8; ++r) {
                    float t = relu1(c[r]);                   // single v_med3
                    if (h == 0) acc0[r] += xv * t;           // o = r + 8*hi
                    else        acc1[r] += xv * t;           // o = 16 + r + 8*hi
                }
            }
        }
        // Compiler fence: keep this group's LDS reads before next group's LDS writes.
        asm volatile("" ::: "memory");

        if (valid) {
            float* srow = summed + (size_t)dstn * OUT_C;
            #pragma unroll
            for (int r = 0; r < 8; ++r) {
                unsafeAtomicAdd(srow + (r + 8 * hi),       acc0[r]);
                unsafeAtomicAdd(srow + (16 + r + 8 * hi),  acc1[r]);
            }
            if (lane < 16) unsafeAtomicAdd(cnt + dstn, 1.0f);
        }
    }
}

// out[n][o] = summed[n][o]/max(cnt[n],1) + sum_i x[n][i]*root[i][o] + bias[o]
__global__ __launch_bounds__(256) void nnconv_finalize_kernel(
    const float* __restrict__ x,
    const float* __restrict__ root,
    const float* __restrict__ bias,
    const float* __restrict__ summed,
    const float* __restrict__ cnt,
    float* __restrict__ out, int N)
{
    int g = blockIdx.x * blockDim.x + threadIdx.x;
    if (g >= N * OUT_C) return;
    int n = g >> 5, o = g & 31;
    float c = fmaxf(cnt[n], 1.0f);
    float acc = summed[g] / c + bias[o];
    const float* xr = x + (size_t)n * IN_C;
    #pragma unroll
    for (int i = 0; i < IN_C; ++i)
        acc += xr[i] * root[i * OUT_C + o];
    out[g] = acc;
}

extern "C" void kernel_launch(void* const* d_in, const int* in_sizes, int n_in,
                              void* d_out, int out_size, void* d_ws, size_t ws_size,
                              hipStream_t stream) {
    const float*     x          = (const float*)d_in[0];
    const long long* edge_index = (const long long*)d_in[1];
    const float*     edge_attr  = (const float*)d_in[2];
    const float*     W_edge     = (const float*)d_in[3];
    const float*     b_edge     = (const float*)d_in[4];
    const float*     root       = (const float*)d_in[5];
    const float*     bias       = (const float*)d_in[6];
    float*           out        = (float*)d_out;

    const int N = in_sizes[0] / IN_C;
    const int E = in_sizes[2] / EDGE_F;

    float* summed = (float*)d_ws;
    float* cnt    = summed + (size_t)N * OUT_C;

    // Zero accumulators every call (deterministic, graph-replay safe).
    int ztot = N * OUT_C + N;
    nnconv_zero_kernel<<<(ztot + 255) / 256, 256, 0, stream>>>(summed, ztot);

    const int G = (E + 15) / 16;
    if (G > 0) {
        int nb = (G + WAVES_PER_BLOCK - 1) / WAVES_PER_BLOCK;
        if (nb > 1024) nb = 1024;
        size_t shmem = (size_t)(WLDS_FLOATS + WAVES_PER_BLOCK * XS_FLOATS) * sizeof(float);
        nnconv_edge_kernel<<<nb, 256, shmem, stream>>>(
            x, edge_index, edge_attr, W_edge, b_edge, summed, cnt, E, G);
    }

    nnconv_finalize_kernel<<<(N * OUT_C + 255) / 256, 256, 0, stream>>>(
        x, root, bias, summed, cnt, out, N);
}